// K_graph_16320875725290
// MI455X (gfx1250) — compile-verified
//
#include <hip/hip_runtime.h>
#include <math.h>

#define BB   1024
#define CC   32
#define HH   64
#define NNF  16
#define NCF  16
#define VV   100
#define KK   8
#define FULLW ((KK + CC) * HH)   // 2560

typedef __attribute__((ext_vector_type(2))) float v2f;
typedef __attribute__((ext_vector_type(8))) float v8f;

// ---------------- reduction helpers (wave32 / 256-thread blocks) ----------------
__device__ __forceinline__ float wave_sum(float v) {
  for (int o = 16; o > 0; o >>= 1) v += __shfl_xor(v, o, 32);
  return v;
}
__device__ __forceinline__ float block_sum256(float v) {
  __shared__ float r[256];
  int t = threadIdx.x; r[t] = v; __syncthreads();
  for (int s = 128; s > 0; s >>= 1) { if (t < s) r[t] += r[t + s]; __syncthreads(); }
  float res = r[0]; __syncthreads();
  return res;
}
__device__ __forceinline__ float block_max256(float v) {
  __shared__ float r[256];
  int t = threadIdx.x; r[t] = v; __syncthreads();
  for (int s = 128; s > 0; s >>= 1) { if (t < s) r[t] = fmaxf(r[t], r[t + s]); __syncthreads(); }
  float res = r[0]; __syncthreads();
  return res;
}

// ---------------- stage 1: raw features + global LN (per half) ----------------
__global__ void k_featraw(const float* __restrict__ num_data, const int* __restrict__ cat_data,
                          const float* __restrict__ num_w, const float* __restrict__ num_b,
                          const float* __restrict__ cat_emb,
                          float* __restrict__ fraw, float* __restrict__ psum, float* __restrict__ psq) {
  int idx = blockIdx.x * 256 + threadIdx.x;
  int b = idx >> 11, j = idx & 2047;
  float v;
  if (j < 1024) {                       // numeric half: relu(x*w + b)
    int n = j >> 6, h = j & 63;
    v = fmaxf(num_data[b * NNF + n] * num_w[n * HH + h] + num_b[n * HH + h], 0.f);
  } else {                              // categorical half: embedding lookup (no relu)
    int jj = j - 1024; int c = jj >> 6, h = jj & 63;
    v = cat_emb[((size_t)c * VV + cat_data[b * NCF + c]) * HH + h];
  }
  fraw[idx] = v;
  float s = block_sum256(v);
  float q = block_sum256(v * v);
  if (threadIdx.x == 0) { psum[blockIdx.x] = s; psq[blockIdx.x] = q; }
}

__global__ void k_ln_reduce(const float* __restrict__ psum, const float* __restrict__ psq,
                            float* __restrict__ scal) {
  float s0 = 0, q0 = 0, s1 = 0, q1 = 0;
  for (int j = threadIdx.x; j < 8192; j += 256) {
    if (((j >> 2) & 1) == 0) { s0 += psum[j]; q0 += psq[j]; }
    else                      { s1 += psum[j]; q1 += psq[j]; }
  }
  s0 = block_sum256(s0); q0 = block_sum256(q0);
  s1 = block_sum256(s1); q1 = block_sum256(q1);
  if (threadIdx.x == 0) {
    const float cnt = 1024.f * 1024.f;
    float mu0 = s0 / cnt, var0 = q0 / cnt - mu0 * mu0;
    float mu1 = s1 / cnt, var1 = q1 / cnt - mu1 * mu1;
    scal[0] = mu0; scal[1] = rsqrtf(var0 + 1e-5f);
    scal[2] = mu1; scal[3] = rsqrtf(var1 + 1e-5f);
  }
}

__global__ void k_ln_apply(const float* __restrict__ fraw, const float* __restrict__ scal,
                           float* __restrict__ fN) {
  int idx = blockIdx.x * 256 + threadIdx.x;
  int j = idx & 2047;
  int h2 = (j >= 1024) ? 2 : 0;
  fN[idx] = (fraw[idx] - scal[h2]) * scal[h2 + 1];
}

// ---------------- generic f32 WMMA GEMM: C[M,N] = A[M,K] @ B(^T) + bias -------
// 4 waves/block; each wave computes a 16(M) x 64(N) strip with 4 accumulators,
// reusing one A fragment across 4 WMMAs per k-step (4 independent XDL chains).
// Block covers 64(M) x 64(N). Requires M%64==0, N%64==0, K%4==0, lda/ldb even.
template<bool BT>
__global__ void k_gemm(const float* __restrict__ A, const float* __restrict__ B,
                       const float* __restrict__ bias, float* __restrict__ C,
                       int M, int N, int K, int lda, int ldb, int ldc) {
  int wave = threadIdx.x >> 5;
  int lane = threadIdx.x & 31;
  int lr = lane & 15, lh = lane >> 4;
  int m0 = blockIdx.y * 64 + wave * 16;
  int n0 = blockIdx.x * 64;
  if (m0 >= M || n0 >= N) return;
  v8f acc0 = {}, acc1 = {}, acc2 = {}, acc3 = {};
  const float* arow = A + (size_t)(m0 + lr) * lda;
  for (int k0 = 0; k0 < K; k0 += 4) {
    int ka = k0 + 2 * lh;              // A 16x4 f32 layout: lane-half selects K pair
    v2f a = *(const v2f*)(arow + ka);  // 8B load, even-aligned
    v2f b0, b1, b2, b3;
    if (BT) {                          // B stored as [N,K] row-major (B^T multiply)
      b0 = *(const v2f*)(B + (size_t)(n0      + lr) * ldb + ka);
      b1 = *(const v2f*)(B + (size_t)(n0 + 16 + lr) * ldb + ka);
      b2 = *(const v2f*)(B + (size_t)(n0 + 32 + lr) * ldb + ka);
      b3 = *(const v2f*)(B + (size_t)(n0 + 48 + lr) * ldb + ka);
    } else {                           // B stored as [K,N] row-major
      const float* br0 = B + (size_t)ka * ldb + n0 + lr;
      const float* br1 = B + (size_t)(ka + 1) * ldb + n0 + lr;
      b0.x = br0[0];  b0.y = br1[0];
      b1.x = br0[16]; b1.y = br1[16];
      b2.x = br0[32]; b2.y = br1[32];
      b3.x = br0[48]; b3.y = br1[48];
    }
    acc0 = __builtin_amdgcn_wmma_f32_16x16x4_f32(false, a, false, b0, (short)0, acc0, false, false);
    acc1 = __builtin_amdgcn_wmma_f32_16x16x4_f32(false, a, false, b1, (short)0, acc1, false, false);
    acc2 = __builtin_amdgcn_wmma_f32_16x16x4_f32(false, a, false, b2, (short)0, acc2, false, false);
    acc3 = __builtin_amdgcn_wmma_f32_16x16x4_f32(false, a, false, b3, (short)0, acc3, false, false);
  }
  v8f accs[4] = {acc0, acc1, acc2, acc3};
#pragma unroll
  for (int t = 0; t < 4; t++) {
    int nc = n0 + t * 16 + lr;
    float bs = bias ? bias[nc] : 0.f;
#pragma unroll
    for (int v = 0; v < 8; v++) {
      int row = m0 + v + 8 * lh;       // C/D layout: VGPR v -> M = v + 8*lanehalf
      C[(size_t)row * ldc + nc] = accs[t][v] + bs;
    }
  }
}

// ---------------- row-wise relu + LN(last dim, gamma/beta), rows of 64 --------
__global__ void k_row_relu_ln(float* __restrict__ x, const float* __restrict__ g,
                              const float* __restrict__ bvec, int rows) {
  int gw = (blockIdx.x * blockDim.x + threadIdx.x) >> 5;
  int lane = threadIdx.x & 31;
  if (gw >= rows) return;
  float* r = x + (size_t)gw * 64;
  float v0 = fmaxf(r[lane], 0.f), v1 = fmaxf(r[lane + 32], 0.f);
  float s = wave_sum(v0 + v1);
  float q = wave_sum(v0 * v0 + v1 * v1);
  float mu = s * (1.f / 64.f);
  float var = q * (1.f / 64.f) - mu * mu;
  float rstd = rsqrtf(var + 1e-5f);
  r[lane]      = (v0 - mu) * rstd * g[lane]      + bvec[lane];
  r[lane + 32] = (v1 - mu) * rstd * g[lane + 32] + bvec[lane + 32];
}

// ---------------- importance: imp_pre = h @ fi_w2 + fi_b2 --------------------
__global__ void k_imp(const float* __restrict__ h, const float* __restrict__ w2,
                      const float* __restrict__ b2, float* __restrict__ imppre) {
  int gw = (blockIdx.x * blockDim.x + threadIdx.x) >> 5;
  int lane = threadIdx.x & 31;
  if (gw >= BB * CC) return;
  const float* r = h + (size_t)gw * 64;
  float p = r[lane] * w2[lane] + r[lane + 32] * w2[lane + 32];
  p = wave_sum(p);
  if (lane == 0) imppre[gw] = p + b2[0];
}

__global__ void k_impln(const float* __restrict__ imppre, float* __restrict__ imp) {
  float s = 0, q = 0;
  for (int t = threadIdx.x; t < BB * CC; t += 256) { float v = imppre[t]; s += v; q += v * v; }
  s = block_sum256(s); q = block_sum256(q);
  __shared__ float sc[2];
  if (threadIdx.x == 0) {
    float cnt = (float)(BB * CC);
    float mu = s / cnt, var = q / cnt - mu * mu;
    sc[0] = mu; sc[1] = rsqrtf(var + 1e-5f);
  }
  __syncthreads();
  for (int t = threadIdx.x; t < BB * CC; t += 256) imp[t] = (imppre[t] - sc[0]) * sc[1];
}

__global__ void k_scalefeat(float* __restrict__ fN, const float* __restrict__ imp) {
  int idx = blockIdx.x * 256 + threadIdx.x;
  int b = idx >> 11, j = idx & 2047;
  fN[idx] *= imp[b * CC + (j >> 6)];
}

// ---------------- top-8 + masked softmax: one wave per row, lane == channel ---
__global__ void k_topk(const float* __restrict__ imp, float* __restrict__ p,
                       unsigned* __restrict__ maskb) {
  int gw = (blockIdx.x * blockDim.x + threadIdx.x) >> 5;
  int lane = threadIdx.x & 31;
  if (gw >= BB) return;
  float v = imp[gw * CC + lane];
  float vv = v;
  int sel = 0;
  for (int k = 0; k < KK; k++) {
    float m = vv; int mi = lane;
    for (int o = 16; o > 0; o >>= 1) {
      float om = __shfl_xor(m, o, 32); int oi = __shfl_xor(mi, o, 32);
      if (om > m || (om == m && oi < mi)) { m = om; mi = oi; }
    }
    if (lane == mi) { sel = 1; vv = -3.0e38f; }
  }
  unsigned mb = (unsigned)__ballot(sel);
  float ms = sel ? v : -3.0e38f;
  for (int o = 16; o > 0; o >>= 1) ms = fmaxf(ms, __shfl_xor(ms, o, 32));
  float e = sel ? expf(v - ms) : 0.f;
  float Z = wave_sum(e);
  p[gw * CC + lane] = e / Z;
  if (lane == 0) maskb[gw] = mb;
}

// ---------------- stable compaction of selected rows per channel -------------
__global__ void k_compact(const unsigned* __restrict__ maskb, int* __restrict__ crow,
                          int* __restrict__ cnt) {
  int c = blockIdx.x;
  int i = threadIdx.x;                  // 1024 threads, one per batch row
  int wid = i >> 5, lane = i & 31;
  int sel = (maskb[i] >> c) & 1;
  unsigned wb = (unsigned)__ballot(sel);
  int wpre = __popc(wb & ((1u << lane) - 1u));
  __shared__ int wtot[32], woff[32];
  if (lane == 0) wtot[wid] = __popc(wb);
  __syncthreads();
  if (i == 0) {
    int acc = 0;
    for (int j = 0; j < 32; j++) { woff[j] = acc; acc += wtot[j]; }
    cnt[c] = acc;
  }
  __syncthreads();
  if (sel) crow[c * 1024 + woff[wid] + wpre] = i;
}

__global__ void k_offsets(const int* __restrict__ cnt, int* __restrict__ coff) {
  if (threadIdx.x == 0) {
    int a = 0;
    for (int c = 0; c < 32; c++) { coff[c] = a; a += cnt[c]; }
    coff[32] = a;                       // always 8192 (each row selects exactly 8)
  }
}

// ---------------- per-channel graph stats: mx, Z, deg^-1/2 -------------------
__global__ void k_graph_stats(const float* __restrict__ D, const float* __restrict__ p,
                              const int* __restrict__ crow, const int* __restrict__ cnt,
                              const int* __restrict__ coff, float* __restrict__ dinv,
                              float* __restrict__ mxv, float* __restrict__ zinvv) {
  int c = blockIdx.x;
  int n = cnt[c], base = coff[c];
  __shared__ int   rows_s[1024];
  __shared__ float pv_s[1024];
  __shared__ float rsum_s[1024];
  for (int t = threadIdx.x; t < n; t += 256) {
    int i = crow[c * 1024 + t];
    rows_s[t] = i; pv_s[t] = p[i * CC + c];
  }
  __syncthreads();
  float lmx = -3.0e38f;
  long nn = (long)n * n;
  for (long t = threadIdx.x; t < nn; t += 256) {
    int a = (int)(t / n), b = (int)(t % n);
    if (a != b) {
      float s = D[(size_t)rows_s[a] * BB + rows_s[b]] - pv_s[a] * pv_s[b];
      if (s > 0.f) lmx = fmaxf(lmx, s);
    }
  }
  float mx = block_max256(lmx);
  if (mx < -1e37f) mx = 0.f;            // no positive entries -> ref uses 0
  for (int a = threadIdx.x; a < n; a += 256) {
    float rs = 0.f;
    int ia = rows_s[a]; float pa = pv_s[a];
    const float* Drow = D + (size_t)ia * BB;
    for (int b = 0; b < n; b++) {
      if (b == a) continue;
      float s = Drow[rows_s[b]] - pa * pv_s[b];
      if (s > 0.f) rs += expf(s - mx);
    }
    rsum_s[a] = rs;
  }
  __syncthreads();
  float lz = 0.f;
  for (int a = threadIdx.x; a < n; a += 256) lz += rsum_s[a];
  float Z = block_sum256(lz);
  float zin = (Z > 0.f) ? (1.f / Z) : 1.f;
  for (int a = threadIdx.x; a < n; a += 256) {
    float deg = 1.f + rsum_s[a] * zin;  // +1 from identity on the support
    dinv[base + a] = rsqrtf(deg);
  }
  if (threadIdx.x == 0) { mxv[c] = mx; zinvv[c] = zin; }
}

// ---------------- x = relu(An @ src + bias), then masked-LN over the block ----
__global__ void k_an_apply(const float* __restrict__ D, const float* __restrict__ p,
                           const int* __restrict__ crow, const int* __restrict__ cnt,
                           const int* __restrict__ coff, const float* __restrict__ dinv,
                           const float* __restrict__ mxv, const float* __restrict__ zinvv,
                           const float* __restrict__ src, int srcFlat,
                           const float* __restrict__ bias, float* __restrict__ dst) {
  int c = blockIdx.x;
  int n = cnt[c], base = coff[c];
  __shared__ int   rows_s[1024];
  __shared__ float pv_s[1024];
  __shared__ float dv_s[1024];
  for (int t = threadIdx.x; t < n; t += 256) {
    int i = crow[c * 1024 + t];
    rows_s[t] = i; pv_s[t] = p[i * CC + c]; dv_s[t] = dinv[base + t];
  }
  __syncthreads();
  float mx = mxv[c], zin = zinvv[c];
  int wave = threadIdx.x >> 5, lane = threadIdx.x & 31;
  float psm = 0.f, psq = 0.f;
  for (int a = wave; a < n; a += 8) {
    float acc0 = 0.f, acc1 = 0.f;
    int ia = rows_s[a]; float pa = pv_s[a], da = dv_s[a];
    const float* Drow = D + (size_t)ia * BB;
    for (int b = 0; b < n; b++) {
      float coef;
      if (b == a) coef = da * da;
      else {
        float s = Drow[rows_s[b]] - pa * pv_s[b];
        coef = (s > 0.f) ? expf(s - mx) * zin * da * dv_s[b] : 0.f;
      }
      if (coef != 0.f) {
        const float* sr = src + (size_t)(srcFlat ? (base + b) : rows_s[b]) * HH;
        acc0 += coef * sr[lane];
        acc1 += coef * sr[lane + 32];
      }
    }
    acc0 = fmaxf(acc0 + bias[lane], 0.f);
    acc1 = fmaxf(acc1 + bias[lane + 32], 0.f);
    dst[(size_t)(base + a) * HH + lane]      = acc0;
    dst[(size_t)(base + a) * HH + lane + 32] = acc1;
    psm += acc0 + acc1; psq += acc0 * acc0 + acc1 * acc1;
  }
  float tot  = block_sum256(psm);
  float totq = block_sum256(psq);
  float cntf = fmaxf((float)n * 64.f, 1.f);
  float mu   = tot / cntf;
  float var  = totq / cntf - mu * mu;
  float rstd = rsqrtf(var + 1e-5f);
  for (int t = threadIdx.x; t < n * HH; t += 256) {
    float* px = dst + (size_t)base * HH + t;
    *px = (*px - mu) * rstd;
  }
}

// ---------------- scatter x2 into full[:, :512] by sorted-channel slot -------
__global__ void k_scatter(const float* __restrict__ x2, const int* __restrict__ crow,
                          const int* __restrict__ cnt, const int* __restrict__ coff,
                          const unsigned* __restrict__ maskb, float* __restrict__ full) {
  int c = blockIdx.x;
  int n = cnt[c], base = coff[c];
  for (int t = threadIdx.x; t < n * HH; t += 256) {
    int a = t >> 6, h = t & 63;
    int i = crow[c * 1024 + a];
    int slot = __popc(maskb[i] & ((1u << c) - 1u));   // rank of c among selected
    full[(size_t)i * FULLW + slot * HH + h] = x2[(size_t)(base + a) * HH + h];
  }
}

__global__ void k_copyfe(const float* __restrict__ fN, float* __restrict__ full) {
  int idx = blockIdx.x * 256 + threadIdx.x;
  int b = idx >> 11, j = idx & 2047;
  full[(size_t)b * FULLW + 512 + j] = fN[idx];
}

// ---------------- final tiny projection to 2 outputs -------------------------
__global__ void k_final(const float* __restrict__ fln, const float* __restrict__ pw2,
                        const float* __restrict__ pb2, float* __restrict__ out) {
  int b = blockIdx.x * 256 + threadIdx.x;
  if (b >= BB) return;
  const float* r = fln + (size_t)b * 64;
  float a0 = 0.f, a1 = 0.f;
  for (int k = 0; k < 64; k++) { a0 += r[k] * pw2[k * 2]; a1 += r[k] * pw2[k * 2 + 1]; }
  out[b * 2]     = a0 + pb2[0];
  out[b * 2 + 1] = a1 + pb2[1];
}

// =============================== host driver =================================
extern "C" void kernel_launch(void* const* d_in, const int* in_sizes, int n_in,
                              void* d_out, int out_size, void* d_ws, size_t ws_size,
                              hipStream_t stream) {
  (void)in_sizes; (void)n_in; (void)out_size; (void)ws_size;
  const float* num_data = (const float*)d_in[0];
  const int*   cat_data = (const int*)  d_in[1];
  const float* num_w    = (const float*)d_in[2];
  const float* num_b    = (const float*)d_in[3];
  const float* cat_emb  = (const float*)d_in[4];
  const float* fi_w1    = (const float*)d_in[5];
  const float* fi_b1    = (const float*)d_in[6];
  const float* fi_g     = (const float*)d_in[7];
  const float* fi_be    = (const float*)d_in[8];
  const float* fi_w2    = (const float*)d_in[9];
  const float* fi_b2    = (const float*)d_in[10];
  const float* gcn1_w   = (const float*)d_in[11];
  const float* gcn1_b   = (const float*)d_in[12];
  const float* gcn2_w   = (const float*)d_in[13];
  const float* gcn2_b   = (const float*)d_in[14];
  const float* pw1      = (const float*)d_in[15];
  const float* pb1      = (const float*)d_in[16];
  const float* pg       = (const float*)d_in[17];
  const float* pbe      = (const float*)d_in[18];
  const float* pw2      = (const float*)d_in[19];
  const float* pb2      = (const float*)d_in[20];
  float* out = (float*)d_out;

  char* w = (char*)d_ws;
  size_t o = 0;
  auto alloc = [&](size_t bytes) -> char* {
    char* ptr = w + o; o += (bytes + 255) & ~(size_t)255; return ptr;
  };
  float*    fraw   = (float*)alloc((size_t)2097152 * 4);  // B x 2048 raw feats
  float*    fN     = (float*)alloc((size_t)2097152 * 4);  // normalized / later scaled
  float*    psum   = (float*)alloc(8192 * 4);
  float*    psq    = (float*)alloc(8192 * 4);
  float*    scal   = (float*)alloc(4 * 4);
  float*    hpre   = fraw;                                // alias: fraw dead after ln_apply
  float*    imppre = (float*)alloc(32768 * 4);
  float*    imp    = (float*)alloc(32768 * 4);
  float*    pmat   = (float*)alloc(32768 * 4);
  unsigned* maskb  = (unsigned*)alloc(1024 * 4);
  int*      crow   = (int*)alloc(32768 * 4);
  int*      cnt    = (int*)alloc(32 * 4);
  int*      coff   = (int*)alloc(33 * 4);
  float*    Dm     = (float*)alloc((size_t)1048576 * 4);  // P @ P^T
  float*    mxv    = (float*)alloc(32 * 4);
  float*    zinv   = (float*)alloc(32 * 4);
  float*    dinv   = (float*)alloc(8192 * 4);
  float*    g1     = (float*)alloc(65536 * 4);            // feat @ gcn1_w
  float*    x1     = (float*)alloc((size_t)524288 * 4);   // compact 8192 x 64
  float*    z2     = (float*)alloc((size_t)524288 * 4);
  float*    x2     = (float*)alloc((size_t)524288 * 4);
  float*    full   = (float*)alloc((size_t)2621440 * 4);  // B x 2560
  float*    fpre   = (float*)alloc(65536 * 4);

  // Stage 1: features + two global LayerNorms
  k_featraw <<<8192, 256, 0, stream>>>(num_data, cat_data, num_w, num_b, cat_emb, fraw, psum, psq);
  k_ln_reduce<<<1, 256, 0, stream>>>(psum, psq, scal);
  k_ln_apply <<<8192, 256, 0, stream>>>(fraw, scal, fN);

  // Stage 2: feature interaction h = LN(relu(fe3 @ fi_w1 + b)), imp = LN_all(h @ fi_w2 + b)
  k_gemm<false><<<dim3(1, 512), 128, 0, stream>>>(fN, fi_w1, fi_b1, hpre, 32768, 64, 64, 64, 64, 64);
  k_row_relu_ln<<<4096, 256, 0, stream>>>(hpre, fi_g, fi_be, 32768);
  k_imp  <<<4096, 256, 0, stream>>>(hpre, fi_w2, fi_b2, imppre);
  k_impln<<<1, 256, 0, stream>>>(imppre, imp);
  k_scalefeat<<<8192, 256, 0, stream>>>(fN, imp);         // fe3 *= imp

  // Stage 3: top-8 + masked softmax, per-channel compaction
  k_topk   <<<128, 256, 0, stream>>>(imp, pmat, maskb);
  k_compact<<<32, 1024, 0, stream>>>(maskb, crow, cnt);
  k_offsets<<<1, 32, 0, stream>>>(cnt, coff);

  // Stage 4: D = P @ P^T  (S[c,i,j] = D[i,j] - p[i,c]p[j,c] on support)
  k_gemm<true><<<dim3(16, 16), 128, 0, stream>>>(pmat, pmat, nullptr, Dm, 1024, 1024, 32, 32, 32, 1024);
  k_graph_stats<<<32, 256, 0, stream>>>(Dm, pmat, crow, cnt, coff, dinv, mxv, zinv);

  // Stage 5: GCN layers in compact space
  k_gemm<false><<<dim3(1, 16), 128, 0, stream>>>(fN, gcn1_w, nullptr, g1, 1024, 64, 2048, 2048, 64, 64);
  k_an_apply<<<32, 256, 0, stream>>>(Dm, pmat, crow, cnt, coff, dinv, mxv, zinv, g1, 0, gcn1_b, x1);
  k_gemm<false><<<dim3(1, 128), 128, 0, stream>>>(x1, gcn2_w, nullptr, z2, 8192, 64, 64, 64, 64, 64);
  k_an_apply<<<32, 256, 0, stream>>>(Dm, pmat, crow, cnt, coff, dinv, mxv, zinv, z2, 1, gcn2_b, x2);

  // Stage 6: gather + concat + final MLP
  k_scatter<<<32, 256, 0, stream>>>(x2, crow, cnt, coff, maskb, full);
  k_copyfe <<<8192, 256, 0, stream>>>(fN, full);
  k_gemm<false><<<dim3(1, 16), 128, 0, stream>>>(full, pw1, pb1, fpre, 1024, 64, 2560, 2560, 64, 64);
  k_row_relu_ln<<<128, 256, 0, stream>>>(fpre, pg, pbe, 1024);
  k_final<<<4, 256, 0, stream>>>(fpre, pw2, pb2, out);
}